// EnergyFlowGNN_23287312679270
// MI455X (gfx1250) — compile-verified
//
#include <hip/hip_runtime.h>
#include <cstddef>

typedef float v2f __attribute__((ext_vector_type(2)));
typedef float v8f __attribute__((ext_vector_type(8)));

// ---------------------------------------------------------------------------
// native f32 atomic add (global_atomic_add_f32, no CAS loop)
// ---------------------------------------------------------------------------
__device__ __forceinline__ void atomAddF(float* p, float v) {
  unsafeAtomicAdd(p, v);
}

// ---------------------------------------------------------------------------
// degree / normalization
// ---------------------------------------------------------------------------
__global__ void init_deg_kernel(unsigned* __restrict__ deg, int n) {
  int i = blockIdx.x * blockDim.x + threadIdx.x;
  if (i < n) deg[i] = 1u;  // self-loop contributes 1
}

__global__ void count_deg_kernel(const int* __restrict__ dst,
                                 unsigned* __restrict__ deg, int E) {
  int i = blockIdx.x * blockDim.x + threadIdx.x;
  if (i < E) atomicAdd(&deg[dst[i]], 1u);
}

__global__ void make_dis_kernel(float* __restrict__ dis, int n) {
  int i = blockIdx.x * blockDim.x + threadIdx.x;
  if (i < n) {
    unsigned d = ((const unsigned*)dis)[i];
    dis[i] = rsqrtf((float)d);  // d >= 1 always (self-loop)
  }
}

// ---------------------------------------------------------------------------
// layer-1 GEMM: h[n,c] = sum_k x[n,k] * W1[k,c], K=5 (too skinny for WMMA)
// ---------------------------------------------------------------------------
__global__ __launch_bounds__(256) void gemm_in_kernel(
    const float* __restrict__ x, const float* __restrict__ W1,
    float* __restrict__ h, int n_nodes) {
  int idx = blockIdx.x * blockDim.x + threadIdx.x;
  if (idx >= n_nodes * 64) return;
  int n = idx >> 6, c = idx & 63;
  const float* xr = x + (size_t)n * 5;
  float s = 0.f;
#pragma unroll
  for (int k = 0; k < 5; ++k) s = fmaf(xr[k], W1[k * 64 + c], s);
  h[idx] = s;
}

// ---------------------------------------------------------------------------
// layer-2 GEMM via fp32 WMMA: C[N,64] = A[N,64] @ W[64,64]
// one wave per 16-row tile, 4 column tiles, K chained in 16 steps of 4
// ---------------------------------------------------------------------------
__global__ __launch_bounds__(256) void gemm64_wmma_kernel(
    const float* __restrict__ A, const float* __restrict__ W,
    float* __restrict__ C, int ntiles) {
  // LDS: pair-K layout  wlds[kp*64 + n] = (W[2kp][n], W[2kp+1][n])
  __shared__ float2 wlds[32 * 64];  // 16 KB
  for (int i = threadIdx.x; i < 32 * 64; i += blockDim.x) {
    int kp = i >> 6, n = i & 63;
    wlds[i] = make_float2(W[(2 * kp) * 64 + n], W[(2 * kp + 1) * 64 + n]);
  }
  __syncthreads();

  const int lane = threadIdx.x & 31;
  const int wave = threadIdx.x >> 5;
  const int m   = lane & 15;    // row (A) / column-in-tile (B,D)
  const int off = lane >> 4;    // 0: K pair {0,1}, 1: K pair {2,3}

  int tile = blockIdx.x * 8 + wave;       // wave-uniform
  if (tile >= ntiles) return;             // whole wave exits; EXEC stays all-1s
  const int node0 = tile * 16;

  const float* arow = A + (size_t)(node0 + m) * 64;

  v8f acc0 = {}, acc1 = {}, acc2 = {}, acc3 = {};
#pragma unroll
  for (int kp = 0; kp < 16; ++kp) {
    const int k = 4 * kp + 2 * off;       // this lane's K pair start
    v2f a;
    a.x = arow[k];
    a.y = arow[k + 1];
    const float2* wp = wlds + (size_t)(2 * kp + off) * 64;  // row pair k/2
    float2 f0 = wp[0 * 16 + m];
    float2 f1 = wp[1 * 16 + m];
    float2 f2 = wp[2 * 16 + m];
    float2 f3 = wp[3 * 16 + m];
    v2f b0 = {f0.x, f0.y}, b1 = {f1.x, f1.y}, b2 = {f2.x, f2.y}, b3 = {f3.x, f3.y};
    acc0 = __builtin_amdgcn_wmma_f32_16x16x4_f32(false, a, false, b0, (short)0, acc0, false, false);
    acc1 = __builtin_amdgcn_wmma_f32_16x16x4_f32(false, a, false, b1, (short)0, acc1, false, false);
    acc2 = __builtin_amdgcn_wmma_f32_16x16x4_f32(false, a, false, b2, (short)0, acc2, false, false);
    acc3 = __builtin_amdgcn_wmma_f32_16x16x4_f32(false, a, false, b3, (short)0, acc3, false, false);
  }

  // D layout: VGPR j holds C[row = j + 8*off][col = m] per column tile
#pragma unroll
  for (int j = 0; j < 8; ++j) {
    float* crow = C + (size_t)(node0 + 8 * off + j) * 64 + m;
    crow[0]  = acc0[j];
    crow[16] = acc1[j];
    crow[32] = acc2[j];
    crow[48] = acc3[j];
  }
}

// scalar tail for n_nodes not multiple of 16 (not used for N=100000)
__global__ void gemm64_tail_kernel(const float* __restrict__ A,
                                   const float* __restrict__ W,
                                   float* __restrict__ C, int row0, int n_nodes) {
  int idx = blockIdx.x * blockDim.x + threadIdx.x;
  int n = row0 + (idx >> 6);
  if (n >= n_nodes) return;
  int c = idx & 63;
  const float* ar = A + (size_t)n * 64;
  float s = 0.f;
  for (int k = 0; k < 64; ++k) s = fmaf(ar[k], W[k * 64 + c], s);
  C[(size_t)n * 64 + c] = s;
}

// ---------------------------------------------------------------------------
// edge scatter-add, width 64: one wave per edge, 2 channels per lane
// ---------------------------------------------------------------------------
__global__ __launch_bounds__(256) void edge_agg64_kernel(
    const int* __restrict__ src, const int* __restrict__ dst,
    const float* __restrict__ dis, const float* __restrict__ h,
    float* __restrict__ agg, int E) {
  int wid  = (int)((blockIdx.x * (unsigned)blockDim.x + threadIdx.x) >> 5);
  int lane = threadIdx.x & 31;
  if (wid >= E) return;
  int s = src[wid], d = dst[wid];
  float w = dis[s] * dis[d];
  const float* hr = h + (size_t)s * 64;
  float* ar = agg + (size_t)d * 64;
  atomAddF(ar + lane, w * hr[lane]);
  atomAddF(ar + lane + 32, w * hr[lane + 32]);
}

// width-1 scatter (layer 3)
__global__ void edge_agg1_kernel(const int* __restrict__ src,
                                 const int* __restrict__ dst,
                                 const float* __restrict__ dis,
                                 const float* __restrict__ z,
                                 float* __restrict__ out, int E) {
  int i = blockIdx.x * blockDim.x + threadIdx.x;
  if (i >= E) return;
  int s = src[i], d = dst[i];
  atomAddF(&out[d], dis[s] * dis[d] * z[s]);
}

// ---------------------------------------------------------------------------
// epilogues: add self-loop term dis^2 * h, bias, optional ReLU (in place)
// ---------------------------------------------------------------------------
__global__ void epilogue64_kernel(float* __restrict__ agg,
                                  const float* __restrict__ h,
                                  const float* __restrict__ dis,
                                  const float* __restrict__ b, int n_nodes,
                                  int do_relu) {
  int idx = blockIdx.x * blockDim.x + threadIdx.x;
  if (idx >= n_nodes * 64) return;
  int n = idx >> 6, c = idx & 63;
  float dn = dis[n];
  float v = agg[idx] + dn * dn * h[idx] + b[c];
  agg[idx] = do_relu ? fmaxf(v, 0.f) : v;
}

__global__ void epilogue1_kernel(float* __restrict__ out,
                                 const float* __restrict__ z,
                                 const float* __restrict__ dis,
                                 const float* __restrict__ b3, int n) {
  int i = blockIdx.x * blockDim.x + threadIdx.x;
  if (i >= n) return;
  float dn = dis[i];
  out[i] = out[i] + dn * dn * z[i] + b3[0];
}

// ---------------------------------------------------------------------------
// layer-3 GEMV: z[n] = sum_k h[n,k] * W3[k]; one wave per node
// ---------------------------------------------------------------------------
__global__ __launch_bounds__(256) void gemv_out_kernel(
    const float* __restrict__ h, const float* __restrict__ W3,
    float* __restrict__ z, int n_nodes) {
  int wid  = (int)((blockIdx.x * (unsigned)blockDim.x + threadIdx.x) >> 5);
  int lane = threadIdx.x & 31;
  if (wid >= n_nodes) return;
  const float* hr = h + (size_t)wid * 64;
  float s = hr[lane] * W3[lane] + hr[lane + 32] * W3[lane + 32];
#pragma unroll
  for (int o = 16; o > 0; o >>= 1) s += __shfl_down(s, o, 32);
  if (lane == 0) z[wid] = s;
}

__global__ void zero_f_kernel(float* __restrict__ p, int n) {
  int i = blockIdx.x * blockDim.x + threadIdx.x;
  if (i < n) p[i] = 0.f;
}

// ---------------------------------------------------------------------------
extern "C" void kernel_launch(void* const* d_in, const int* in_sizes, int n_in,
                              void* d_out, int out_size, void* d_ws, size_t ws_size,
                              hipStream_t stream) {
  const float* x  = (const float*)d_in[0];
  const int*   ei = (const int*)d_in[1];
  const float* W1 = (const float*)d_in[2];
  const float* b1 = (const float*)d_in[3];
  const float* W2 = (const float*)d_in[4];
  const float* b2 = (const float*)d_in[5];
  const float* W3 = (const float*)d_in[6];
  const float* b3 = (const float*)d_in[7];
  float* out = (float*)d_out;

  const int N = in_sizes[0] / 5;   // 100000
  const int E = in_sizes[1] / 2;   // 3200000
  const int* src = ei;             // edge_index[0]
  const int* dst = ei + E;         // edge_index[1]

  // workspace carve (all 256B-aligned)
  float* dis = (float*)d_ws;
  float* h   = dis + (((size_t)N + 63) & ~(size_t)63);
  float* agg = h + (size_t)N * 64;
  float* z   = agg + (size_t)N * 64;

  const int T = 256;
  const int gN   = (N + T - 1) / T;
  const int gE   = (E + T - 1) / T;
  const int gNC  = (N * 64 + T - 1) / T;
  const int gEw  = (E * 32 + T - 1) / T;      // one wave per edge
  const int gNw  = (N * 32 + T - 1) / T;      // one wave per node
  const int ntiles = N / 16;
  const int gT   = (ntiles + 7) / 8;          // 8 waves/block

  // ---- symmetric normalization: dis = deg^{-1/2}
  init_deg_kernel<<<gN, T, 0, stream>>>((unsigned*)dis, N);
  count_deg_kernel<<<gE, T, 0, stream>>>(dst, (unsigned*)dis, E);
  make_dis_kernel<<<gN, T, 0, stream>>>(dis, N);

  // ---- layer 1
  gemm_in_kernel<<<gNC, T, 0, stream>>>(x, W1, h, N);
  zero_f_kernel<<<gNC, T, 0, stream>>>(agg, N * 64);
  edge_agg64_kernel<<<gEw, T, 0, stream>>>(src, dst, dis, h, agg, E);
  epilogue64_kernel<<<gNC, T, 0, stream>>>(agg, h, dis, b1, N, 1);  // relu1 in agg

  // ---- layer 2 (fp32 WMMA GEMM)
  if (gT > 0)
    gemm64_wmma_kernel<<<gT, T, 0, stream>>>(agg, W2, h, ntiles);
  const int rem = N - ntiles * 16;
  if (rem > 0)
    gemm64_tail_kernel<<<(rem * 64 + T - 1) / T, T, 0, stream>>>(agg, W2, h, ntiles * 16, N);
  zero_f_kernel<<<gNC, T, 0, stream>>>(agg, N * 64);
  edge_agg64_kernel<<<gEw, T, 0, stream>>>(src, dst, dis, h, agg, E);
  epilogue64_kernel<<<gNC, T, 0, stream>>>(agg, h, dis, b2, N, 1);  // relu2 in agg

  // ---- layer 3
  gemv_out_kernel<<<gNw, T, 0, stream>>>(agg, W3, z, N);
  zero_f_kernel<<<gN, T, 0, stream>>>(out, N);
  edge_agg1_kernel<<<gE, T, 0, stream>>>(src, dst, dis, z, out, E);
  epilogue1_kernel<<<gN, T, 0, stream>>>(out, z, dis, b3, N);
}